// Model_Net_13529146983055
// MI455X (gfx1250) — compile-verified
//
#include <hip/hip_runtime.h>
#include <hip/hip_bf16.h>

typedef __attribute__((ext_vector_type(2))) float v2f;
typedef __attribute__((ext_vector_type(8))) float v8f;

#define LN 3
#define NN 100000
#define EE 1600000
#define BB 262144
#define BN_EPS 1e-5f

// ---------------- utility ----------------
__global__ void k_fill(float* __restrict__ p, float v, int n) {
  int i = blockIdx.x * blockDim.x + threadIdx.x;
  if (i < n) p[i] = v;
}

// deg[dst] += 1 per edge (deg pre-filled with 1.0 for self-loop)
__global__ void k_deg(const int* __restrict__ dst, float* __restrict__ deg, int E) {
  int e = blockIdx.x * blockDim.x + threadIdx.x;
  if (e < E) atomicAdd(&deg[dst[e]], 1.0f);
}

__global__ void k_findeg(const float* __restrict__ deg, float* __restrict__ isd,
                         float* __restrict__ invd, int n) {
  int i = blockIdx.x * blockDim.x + threadIdx.x;
  if (i < n) {
    float d = deg[i];
    isd[i]  = rsqrtf(d);
    invd[i] = 1.0f / d;
  }
}

// scalar message pass: s[dst] += x[src] * isd[src]   (rank-1 collapse of conv1)
__global__ void k_smp(const int* __restrict__ ei, const float* __restrict__ x,
                      const float* __restrict__ isd, float* __restrict__ sbuf, int E) {
  int e = blockIdx.x * blockDim.x + threadIdx.x;
  if (e >= E) return;
  int s = ei[e];
  int d = ei[E + e];
  atomicAdd(&sbuf[d], x[s] * isd[s]);
}

// t[n] = isd[n]*s[n] + x[n]*invdeg[n]; accumulate sum(t), sum(t^2) via LDS tree
__global__ void k_tstats(const float* __restrict__ x, const float* __restrict__ isd,
                         const float* __restrict__ invd, float* __restrict__ sbuf,
                         float* __restrict__ stats, int n) {
  __shared__ float sh0[256];
  __shared__ float sh1[256];
  int tid = threadIdx.x;
  int i = blockIdx.x * blockDim.x + tid;
  float t = 0.0f;
  if (i < n) {
    t = isd[i] * sbuf[i] + x[i] * invd[i];
    sbuf[i] = t;  // sbuf now holds t[n]
  }
  sh0[tid] = t;
  sh1[tid] = t * t;
  __syncthreads();
  for (int s = 128; s > 0; s >>= 1) {
    if (tid < s) { sh0[tid] += sh0[tid + s]; sh1[tid] += sh1[tid + s]; }
    __syncthreads();
  }
  if (tid == 0) {
    atomicAdd(&stats[0], sh0[0]);
    atomicAdd(&stats[1], sh1[0]);
  }
}

// fold BN into per-feature scale: sc[k] = gamma*w1*rsqrt(w1^2*var(t)+eps); stats[2]=mean(t)
__global__ void k_sc(const float* __restrict__ stats, const float* __restrict__ w1,
                     const float* __restrict__ gamma, float* __restrict__ sc, int n) {
  int k = threadIdx.x;  // 64 threads, 1 block
  float mt = stats[0] / (float)n;
  float vt = stats[1] / (float)n - mt * mt;
  float w = w1[k];
  sc[k] = gamma[k] * w * rsqrtf(w * w * vt + BN_EPS);
  if (k == 0) ((float*)stats)[2] = mt;
}

// h1[n,k] = relu((t[n]-mt)*sc[k] + beta[k])
__global__ void k_h1(const float* __restrict__ tbuf, const float* __restrict__ stats,
                     const float* __restrict__ sc, const float* __restrict__ beta,
                     float* __restrict__ h1, int total) {
  int idx = blockIdx.x * blockDim.x + threadIdx.x;
  if (idx >= total) return;
  int n = idx >> 6;
  int k = idx & 63;
  float v = (tbuf[n] - stats[2]) * sc[k] + beta[k];
  h1[idx] = fmaxf(v, 0.0f);
}

// 64-wide message pass: S1[dst,:] += h1[src,:] * isd[src]
// 16 threads per edge, 4 features each (float4 gather, 4 f32 atomics)
__global__ void k_mp64(const int* __restrict__ ei, const float* __restrict__ h1,
                       const float* __restrict__ isd, float* __restrict__ S1, int E) {
  int t = blockIdx.x * blockDim.x + threadIdx.x;
  int e = t >> 4;
  int jg = t & 15;
  if (e >= E) return;
  if (jg == 0 && e + 4096 < E) {
    __builtin_prefetch(ei + e + 4096, 0, 1);       // -> global_prefetch_b8
    __builtin_prefetch(ei + E + e + 4096, 0, 1);
  }
  int s = ei[e];
  int d = ei[E + e];
  float w = isd[s];
  float4 hv = ((const float4*)h1)[s * 16 + jg];
  float* o = S1 + (size_t)d * 64 + jg * 4;
  atomicAdd(o + 0, hv.x * w);
  atomicAdd(o + 1, hv.y * w);
  atomicAdd(o + 2, hv.z * w);
  atomicAdd(o + 3, hv.w * w);
}

// embeds = (isd.*S1 + invdeg.*h1) @ W2 + b2  -- f32 WMMA, one wave per 16x16 tile.
// A 16x4 f32 layout: lanes 0-15 hold {K=0,K=1}, lanes 16-31 hold {K=2,K=3} (row M=lane%16).
// B 4x16 mirrored; C/D: VGPR r -> (M=r+8*half, N=lane%16).
__global__ void k_gemm(const float* __restrict__ h1, const float* __restrict__ S1,
                       const float* __restrict__ isd, const float* __restrict__ invd,
                       const float* __restrict__ W2, const float* __restrict__ b2,
                       float* __restrict__ out) {
  int wave = threadIdx.x >> 5;            // 8 waves per block
  int lane = threadIdx.x & 31;
  int tile = blockIdx.x * 8 + wave;       // 50000 tiles total
  int tm = tile >> 3;                     // 0..6249
  int tn = tile & 7;                      // 0..7
  int m0 = tm * 16, n0 = tn * 16;
  int half = lane >> 4, l16 = lane & 15;

  int arow = m0 + l16;
  float sd = isd[arow];
  float iv = invd[arow];
  const float* hrow = h1 + (size_t)arow * 64;
  const float* srow = S1 + (size_t)arow * 64;

  v8f c = {0.f, 0.f, 0.f, 0.f, 0.f, 0.f, 0.f, 0.f};
#pragma unroll
  for (int k0 = 0; k0 < 64; k0 += 4) {
    int ka = k0 + 2 * half;
    v2f a, b;
    a.x = sd * srow[ka]     + iv * hrow[ka];        // fused A-operand build
    a.y = sd * srow[ka + 1] + iv * hrow[ka + 1];
    b.x = W2[ka * 128 + n0 + l16];
    b.y = W2[(ka + 1) * 128 + n0 + l16];
    c = __builtin_amdgcn_wmma_f32_16x16x4_f32(false, a, false, b, (short)0, c,
                                              false, false);
  }
  float bias = b2[n0 + l16];
#pragma unroll
  for (int r = 0; r < 8; ++r) {
    int row = m0 + r + 8 * half;
    out[(size_t)row * 128 + n0 + l16] = c[r] + bias;
  }
}

// link head: one wave per pair; lanes 0-15 -> left embed, 16-31 -> right embed.
__global__ void k_link(const float* __restrict__ embeds, const int* __restrict__ nl,
                       const int* __restrict__ ln, const int* __restrict__ rn,
                       const float* __restrict__ linkW, const float* __restrict__ linkb,
                       float* __restrict__ out) {
  int gid = blockIdx.x * blockDim.x + threadIdx.x;
  int pair = gid >> 5;
  int lane = gid & 31;
  if (pair >= BB) return;
  int layer = nl[pair];
  const float* base = embeds + (size_t)layer * NN * 128;
  int node = (lane < 16) ? ln[pair] : rn[pair];
  const float* vec = base + (size_t)node * 128 + (lane & 15) * 8;
  float4 v0 = *(const float4*)(vec);
  float4 v1 = *(const float4*)(vec + 4);
  int g0 = lane * 8;  // feature index in the concatenated 256-vector
  float4 wa0 = *(const float4*)(linkW + g0);
  float4 wb0 = *(const float4*)(linkW + g0 + 4);
  float4 wa1 = *(const float4*)(linkW + 256 + g0);
  float4 wb1 = *(const float4*)(linkW + 256 + g0 + 4);
  float p0 = v0.x * wa0.x + v0.y * wa0.y + v0.z * wa0.z + v0.w * wa0.w +
             v1.x * wb0.x + v1.y * wb0.y + v1.z * wb0.z + v1.w * wb0.w;
  float p1 = v0.x * wa1.x + v0.y * wa1.y + v0.z * wa1.z + v0.w * wa1.w +
             v1.x * wb1.x + v1.y * wb1.y + v1.z * wb1.z + v1.w * wb1.w;
#pragma unroll
  for (int off = 16; off > 0; off >>= 1) {
    p0 += __shfl_xor(p0, off, 32);
    p1 += __shfl_xor(p1, off, 32);
  }
  if (lane == 0) {
    out[(size_t)pair * 2 + 0] = p0 + linkb[0];
    out[(size_t)pair * 2 + 1] = p1 + linkb[1];
  }
}

// ---------------- host ----------------
extern "C" void kernel_launch(void* const* d_in, const int* in_sizes, int n_in,
                              void* d_out, int out_size, void* d_ws, size_t ws_size,
                              hipStream_t stream) {
  const float* x_feats = (const float*)d_in[0];   // [L,N,1]
  const float* W1      = (const float*)d_in[1];   // [L,1,64]
  const float* b1      = (const float*)d_in[2];   // [L,64] (zeros; folded via BN mean)
  const float* gamma   = (const float*)d_in[3];   // [L,64]
  const float* beta    = (const float*)d_in[4];   // [L,64]
  const float* W2      = (const float*)d_in[5];   // [L,64,128]
  const float* b2      = (const float*)d_in[6];   // [L,128]
  const float* linkW   = (const float*)d_in[7];   // [2,256]
  const float* linkb   = (const float*)d_in[8];   // [2]
  const int* edge_index = (const int*)d_in[9];    // [L,2,E]
  const int* now_layer  = (const int*)d_in[10];   // [B]
  const int* leftnode   = (const int*)d_in[11];   // [B]
  const int* rightnode  = (const int*)d_in[12];   // [B]
  float* out = (float*)d_out;
  (void)b1; (void)in_sizes; (void)n_in; (void)out_size; (void)ws_size;
  // NOTE: b1 shifts the pre-BN activations uniformly per feature, and BN's mean
  // subtraction removes any per-feature constant shift, so b1 cancels exactly.

  float* embeds = (float*)d_ws;                        // L*N*128
  float* h1   = embeds + (size_t)LN * NN * 128;        // N*64
  float* S1   = h1 + (size_t)NN * 64;                  // N*64
  float* isd  = S1 + (size_t)NN * 64;                  // N
  float* invd = isd + NN;                              // N
  float* sbuf = invd + NN;                             // N (deg -> s -> t)
  float* stats = sbuf + NN;                            // 8
  float* sc   = stats + 8;                             // 64

  const int T = 256;
  const int gN  = (NN + T - 1) / T;

  for (int l = 0; l < LN; ++l) {
    const int* ei = edge_index + (size_t)l * 2 * EE;
    const float* x = x_feats + (size_t)l * NN;

    // degrees (with self-loop)
    k_fill<<<gN, T, 0, stream>>>(sbuf, 1.0f, NN);
    k_deg<<<EE / T, T, 0, stream>>>(ei + EE, sbuf, EE);
    k_findeg<<<gN, T, 0, stream>>>(sbuf, isd, invd, NN);

    // rank-1 conv1 collapsed to scalar scatter
    k_fill<<<gN, T, 0, stream>>>(sbuf, 0.0f, NN);
    k_smp<<<EE / T, T, 0, stream>>>(ei, x, isd, sbuf, EE);

    // BN stats on scalar t
    k_fill<<<1, T, 0, stream>>>(stats, 0.0f, 8);
    k_tstats<<<gN, T, 0, stream>>>(x, isd, invd, sbuf, stats, NN);
    k_sc<<<1, 64, 0, stream>>>(stats, W1 + l * 64, gamma + l * 64, sc, NN);

    // h1 = relu(BN(t ⊗ w1))
    k_h1<<<(NN * 64) / T, T, 0, stream>>>(sbuf, stats, sc, beta + l * 64, h1, NN * 64);

    // 64-wide neighbor aggregation of h1 (hoisted before the GEMM)
    k_fill<<<(NN * 64) / T, T, 0, stream>>>(S1, 0.0f, NN * 64);
    k_mp64<<<(EE * 16) / T, T, 0, stream>>>(ei, h1, isd, S1, EE);

    // embeds[l] = (isd.*S1 + invdeg.*h1) @ W2 + b2   (f32 WMMA)
    k_gemm<<<NN / 16, T, 0, stream>>>(h1, S1, isd, invd,
                                      W2 + (size_t)l * 64 * 128, b2 + l * 128,
                                      embeds + (size_t)l * NN * 128);
  }

  // link prediction head
  k_link<<<(BB * 32) / T, T, 0, stream>>>(embeds, now_layer, leftnode, rightnode,
                                          linkW, linkb, out);
}